// GraphSAGELayer_17257178596104
// MI455X (gfx1250) — compile-verified
//
#include <hip/hip_runtime.h>
#include <hip/hip_bf16.h>

#define D 128
#define TWO_D 256

typedef __attribute__((ext_vector_type(2))) float v2f;
typedef __attribute__((ext_vector_type(8))) float v8f;

// ---------------------------------------------------------------------------
// Kernel 1: Pre-LayerNorm.  One wave (32 lanes) per node; each lane owns 4
// contiguous features (float4).  Cross-lane reduction via shfl_xor (wave32).
// ---------------------------------------------------------------------------
__global__ __launch_bounds__(256) void ln_kernel(
    const float* __restrict__ X, const float* __restrict__ gamma,
    const float* __restrict__ beta, float* __restrict__ H, int N) {
  int node = (blockIdx.x * blockDim.x + threadIdx.x) >> 5;
  int lane = threadIdx.x & 31;
  if (node >= N) return;

  const float4 v = ((const float4*)(X + (size_t)node * D))[lane];
  float s = v.x + v.y + v.z + v.w;
#pragma unroll
  for (int off = 16; off > 0; off >>= 1) s += __shfl_xor(s, off, 32);
  const float mu = s * (1.0f / (float)D);

  float dx = v.x - mu, dy = v.y - mu, dz = v.z - mu, dw = v.w - mu;
  float q = dx * dx + dy * dy + dz * dz + dw * dw;
#pragma unroll
  for (int off = 16; off > 0; off >>= 1) q += __shfl_xor(q, off, 32);
  const float rstd = rsqrtf(q * (1.0f / (float)D) + 1e-5f);

  const float4 g = ((const float4*)gamma)[lane];
  const float4 be = ((const float4*)beta)[lane];
  float4 o;
  o.x = dx * rstd * g.x + be.x;
  o.y = dy * rstd * g.y + be.y;
  o.z = dz * rstd * g.z + be.z;
  o.w = dw * rstd * g.w + be.w;
  ((float4*)(H + (size_t)node * D))[lane] = o;
}

// ---------------------------------------------------------------------------
// Kernel 2: edge aggregation  neigh[dst] += w * H[src].
// One wave per edge; lane owns 4 features.  f32 HW atomics (no return ->
// STOREcnt path); neigh (51MB) is L2-resident on the 192MB L2.
// ---------------------------------------------------------------------------
__global__ __launch_bounds__(256) void agg_kernel(
    const float* __restrict__ H, const float* __restrict__ ew,
    const int* __restrict__ esrc, const int* __restrict__ edst,
    float* __restrict__ neigh, int E) {
  int e = (blockIdx.x * blockDim.x + threadIdx.x) >> 5;
  int lane = threadIdx.x & 31;
  if (e >= E) return;

  const int s = esrc[e];
  const int d = edst[e];
  const float w = ew[e];

  const float4 h = ((const float4*)(H + (size_t)s * D))[lane];
  float* np = neigh + (size_t)d * D + lane * 4;
  unsafeAtomicAdd(np + 0, w * h.x);
  unsafeAtomicAdd(np + 1, w * h.y);
  unsafeAtomicAdd(np + 2, w * h.z);
  unsafeAtomicAdd(np + 3, w * h.w);
}

// ---------------------------------------------------------------------------
// Kernel 3: out = relu([H | neigh] @ W + b) + X  via V_WMMA_F32_16X16X4_F32.
// Block = 256 threads = 8 waves.  Block owns 16 rows (nodes); wave w owns the
// 16-column tile [16w, 16w+16).  W (256x128 f32 = 128KB) staged in LDS once.
// K = 256 -> 64 wmma per wave (32 over H, 32 over neigh).
//
// f32 WMMA VGPR layouts (ISA 7.12.2):
//   A 16x4 : lane<16 -> row M=lane,   a = {K=k0,  K=k0+1}
//            lane>=16-> row M=lane-16,a = {K=k0+2,K=k0+3}
//   B 4x16 : same with N=lane (&15)
//   C/D    : vgpr r, lane<16 -> (M=r,   N=lane); lane>=16 -> (M=r+8, N=lane-16)
// ---------------------------------------------------------------------------
__global__ __launch_bounds__(256) void gemm_kernel(
    const float* __restrict__ H, const float* __restrict__ neigh,
    const float* __restrict__ W, const float* __restrict__ bias,
    const float* __restrict__ X, float* __restrict__ out, int N) {
  extern __shared__ float smem[];          // 256*128 W + 128 bias
  float* sW = smem;
  float* sB = smem + TWO_D * D;

  // Cooperative stage of W into LDS (float4 copies), bias by first 32 threads.
  {
    const float4* src4 = (const float4*)W;
    float4* dst4 = (float4*)sW;
#pragma unroll
    for (int i = 0; i < (TWO_D * D / 4) / 256; ++i)
      dst4[threadIdx.x + i * 256] = src4[threadIdx.x + i * 256];
    if (threadIdx.x < D / 4)
      ((float4*)sB)[threadIdx.x] = ((const float4*)bias)[threadIdx.x];
  }
  __syncthreads();

  const int wave = threadIdx.x >> 5;       // 0..7 -> column tile
  const int lane = threadIdx.x & 31;
  const int lhalf = lane >> 4;             // 0: K+{0,1}, 1: K+{2,3}
  const int lrow = lane & 15;              // M (for A) / N (for B)
  const int row0 = blockIdx.x * 16;
  const int col0 = wave * 16;

  int arow = row0 + lrow;
  if (arow >= N) arow = N - 1;             // clamp reads; EXEC stays all-1s
  const float* aH = H + (size_t)arow * D;
  const float* aN = neigh + (size_t)arow * D;
  const float* bp = sW + col0 + lrow;      // index with k*D

  v8f c = {0.f, 0.f, 0.f, 0.f, 0.f, 0.f, 0.f, 0.f};

#pragma unroll
  for (int kk = 0; kk < 32; ++kk) {        // K in [0,128): H half of cat
    const int k0 = kk * 4 + lhalf * 2;
    const float2 av = *(const float2*)(aH + k0);
    v2f a;  a.x = av.x;          a.y = av.y;
    v2f b;  b.x = bp[(size_t)k0 * D];  b.y = bp[(size_t)(k0 + 1) * D];
    c = __builtin_amdgcn_wmma_f32_16x16x4_f32(false, a, false, b, (short)0, c,
                                              false, false);
  }
#pragma unroll
  for (int kk = 0; kk < 32; ++kk) {        // K in [128,256): neigh half
    const int k0 = kk * 4 + lhalf * 2;
    const float2 av = *(const float2*)(aN + k0);
    v2f a;  a.x = av.x;                    a.y = av.y;
    v2f b;  b.x = bp[(size_t)(D + k0) * D];  b.y = bp[(size_t)(D + k0 + 1) * D];
    c = __builtin_amdgcn_wmma_f32_16x16x4_f32(false, a, false, b, (short)0, c,
                                              false, false);
  }

  // Epilogue: bias + relu + residual.  All WMMAs done -> divergence is OK now.
  const int n = lrow;
  const float bv = sB[col0 + n];
#pragma unroll
  for (int r = 0; r < 8; ++r) {
    const int m = r + lhalf * 8;
    const int row = row0 + m;
    if (row < N) {
      const size_t idx = (size_t)row * D + col0 + n;
      float v = c[r] + bv;
      v = v > 0.f ? v : 0.f;
      out[idx] = v + X[idx];
    }
  }
}

// ---------------------------------------------------------------------------
extern "C" void kernel_launch(void* const* d_in, const int* in_sizes, int n_in,
                              void* d_out, int out_size, void* d_ws,
                              size_t ws_size, hipStream_t stream) {
  const float* X     = (const float*)d_in[0];
  const float* ew    = (const float*)d_in[1];
  const float* W     = (const float*)d_in[2];
  const float* b     = (const float*)d_in[3];
  const float* gamma = (const float*)d_in[4];
  const float* beta  = (const float*)d_in[5];
  const int*   esrc  = (const int*)d_in[6];
  const int*   edst  = (const int*)d_in[7];
  float* out = (float*)d_out;

  const int N = in_sizes[0] / D;   // 100000
  const int E = in_sizes[1];       // 1600000

  float* Hbuf  = (float*)d_ws;                 // N*D floats
  float* neigh = Hbuf + (size_t)N * D;         // N*D floats

  // zero the atomic accumulator (stream-ordered, graph-capture safe)
  hipMemsetAsync(neigh, 0, (size_t)N * D * sizeof(float), stream);

  // 1) LayerNorm: 8 nodes (waves) per 256-thread block
  ln_kernel<<<(N + 7) / 8, 256, 0, stream>>>(X, gamma, beta, Hbuf, N);

  // 2) Edge scatter-add: 8 edges per block
  agg_kernel<<<(E + 7) / 8, 256, 0, stream>>>(Hbuf, ew, esrc, edst, neigh, E);

  // 3) WMMA GEMM + bias + relu + residual: one 16-row block per workgroup
  const size_t smem = (TWO_D * D + D) * sizeof(float);
  gemm_kernel<<<(N + 15) / 16, 256, smem, stream>>>(Hbuf, neigh, W, b, X, out,
                                                    N);
}